// OrientedAttention_66589172957776
// MI455X (gfx1250) — compile-verified
//
#include <hip/hip_runtime.h>

typedef float v2f __attribute__((ext_vector_type(2)));
typedef float v8f __attribute__((ext_vector_type(8)));
typedef int   vi4 __attribute__((vector_size(16)));   // matches builtin param type

#define LQ 1000          // queries per batch
#define BQ 2             // batch
#define NBL (BQ * LQ)    // 2000 query rows
#define KC 32            // K-chunk staged in LDS per iteration

#define AS1 __attribute__((address_space(1)))
#define AS3 __attribute__((address_space(3)))

// gfx1250 async global->LDS DMA if the toolchain exposes it; sync fallback else.
#if defined(__AMDGCN__) && __has_builtin(__builtin_amdgcn_global_load_async_to_lds_b128)
  #define HAVE_ASYNC_LDS 1
#else
  #define HAVE_ASYNC_LDS 0
#endif

#if defined(__AMDGCN__) && __has_builtin(__builtin_amdgcn_s_wait_asynccnt)
  #define WAIT_ASYNC() __builtin_amdgcn_s_wait_asynccnt(0)
#else
  #define WAIT_ASYNC() asm volatile("s_wait_asynccnt 0x0" ::: "memory")
#endif

// ---------------------------------------------------------------------------
// (B,C,H,W) -> (B,H,W,C) transpose so bilinear gathers become coalesced 256B
// reads (entire pyramid = 178MB, resident in MI455X's 192MB L2 afterwards).
// ---------------------------------------------------------------------------
__global__ void transpose_chw_hwc(const float* __restrict__ src,
                                  float* __restrict__ dst, int HW) {
  const int hw = blockIdx.x;
  const int b  = blockIdx.y;
  const int c  = threadIdx.x;               // 256 channels
  dst[((size_t)b * HW + hw) * 256 + c] =
      src[((size_t)b * 256 + c) * HW + hw];
}

// ---------------------------------------------------------------------------
// Exact-fp32 WMMA GEMM: C[M,N] = act(A[M,K] @ B[K,N] + bias) (+ resid)
// Block = 128 threads = 4 waves sharing one 16-row A strip via LDS
// (async-DMA staged when available); each wave owns a 16x16 N-tile and steps
// K by 4 with V_WMMA_F32_16X16X4_F32. All loop addresses are strength-reduced
// to running pointers. M,K,N are exact multiples here -> EXEC all-ones at WMMA.
// act: 0=none, 1=relu, 2=sigmoid
// ---------------------------------------------------------------------------
__global__ void wmma_gemm_f32(const float* __restrict__ A,
                              const float* __restrict__ Bm,
                              const float* __restrict__ bias,
                              const float* __restrict__ resid,
                              float* __restrict__ C,
                              int M, int N, int K, int act) {
  __shared__ float As[16 * KC];             // 2 KB

  const int tid  = threadIdx.x;
  const int wave = tid >> 5;
  const int lane = tid & 31;
  const int mr   = lane & 15;               // row (A) / col (B,C) within tile
  const int half = lane >> 4;               // selects K pair {0,1} vs {2,3}

  const int m0 = blockIdx.x * 16;
  const int n0 = (blockIdx.y * 4 + wave) * 16;
  const int n  = n0 + mr;

  // cooperative A staging: 128 threads x float4 == 16 x 32 tile
  const int arow = tid >> 3;                // 0..15
  const int acol = (tid & 7) * 4;           // 0,4,...,28
  const float* ag = A + (size_t)(m0 + arow) * K + acol;

  // per-wave B walker: rows {2*half, 2*half+1} of current chunk, column n
  const float* bp = Bm + (size_t)(2 * half) * N + n;

  v8f acc = {};

  for (int kc = 0; kc < K; kc += KC) {
#if HAVE_ASYNC_LDS
    __builtin_amdgcn_global_load_async_to_lds_b128(
        (AS1 vi4*)ag, (AS3 vi4*)&As[arow * KC + acol], 0, 0);
    WAIT_ASYNC();
#else
    *(float4*)&As[arow * KC + acol] = *(const float4*)ag;
#endif
    ag += KC;
    __syncthreads();

    // pull next B chunk toward L2 while this one computes
    __builtin_prefetch(bp + (size_t)KC * N, 0, 1);

    int la = mr * KC + 2 * half;
#pragma unroll
    for (int kk = 0; kk < KC; kk += 4) {
      v2f a = *(const v2f*)&As[la];          // ds_load_b64
      v2f b;
      b.x = bp[0];
      b.y = bp[N];
      acc = __builtin_amdgcn_wmma_f32_16x16x4_f32(
          /*neg_a=*/false, a, /*neg_b=*/false, b,
          /*c_mod=*/(short)0, acc, /*reuse_a=*/false, /*reuse_b=*/false);
      la += 4;
      bp += (size_t)4 * N;
    }
    __syncthreads();
  }

  const float bv = bias ? bias[n] : 0.0f;
#pragma unroll
  for (int r = 0; r < 8; ++r) {
    const int m = m0 + half * 8 + r;        // lanes 0-15: rows 0-7; 16-31: rows 8-15
    float v = acc[r] + bv;
    if (act == 1)      v = fmaxf(v, 0.0f);
    else if (act == 2) v = 1.0f / (1.0f + __expf(-v));
    if (resid) v += resid[(size_t)m * N + n];
    C[(size_t)m * N + n] = v;
  }
}

// ---------------------------------------------------------------------------
// Oriented deformable sampling + channel/spatial modulation.
// Block = 128 threads = 4 waves, one wave per head; block per query (b,l).
// Each lane owns 2 channels -> every corner gather is one coalesced float2
// (wave reads 256B contiguous from the HWC-transposed level, L2-resident).
// Writes vmod[b,l, (g*32+p)*64 + c] ready for the W_out GEMM.
// ---------------------------------------------------------------------------
__global__ void sample_modulate(const float* __restrict__ vt0,
                                const float* __restrict__ vt1,
                                const float* __restrict__ vt2,
                                const float* __restrict__ vt3,
                                const float* __restrict__ qxyzrt,
                                const float* __restrict__ offs,
                                const float* __restrict__ ca,
                                const float* __restrict__ sa,
                                float* __restrict__ vmod) {
  const int bl   = blockIdx.x;            // b*L + l
  const int g    = threadIdx.x >> 5;      // head (wave id)
  const int lane = threadIdx.x & 31;
  const int b    = bl / LQ;

  const float* q = qxyzrt + (size_t)bl * 5;
  const float cx = q[0], cy = q[1], z = q[2], r = q[3], th = q[4];
  const float scale = exp2f(z);
  const float rw = scale * exp2f(-0.5f * r);   // roi width
  const float rh = scale * exp2f( 0.5f * r);   // roi height
  const float cth = cosf(th), sth = sinf(th);

  const float* vts[4]  = {vt0, vt1, vt2, vt3};
  const int    Hs[4]   = {256, 128, 64, 32};
  const float  strd[4] = {4.f, 8.f, 16.f, 32.f};
  const float  glog[4] = {2.f, 3.f, 4.f, 5.f};   // log2(strides)

  const float cav0 = ca[(size_t)bl * 256 + g * 64 + 2 * lane];
  const float cav1 = ca[(size_t)bl * 256 + g * 64 + 2 * lane + 1];

  for (int p = 0; p < 32; ++p) {
    const int idx = g * 32 + p;
    const float* o = offs + (size_t)bl * 384 + idx * 3;
    const float ox = o[0] * rw, oy = o[1] * rh;
    const float sx = cx + ox * cth - oy * sth;   // R @ off_xy
    const float sy = cy + ox * sth + oy * cth;
    const float sz = z + o[2];

    float a0 = 0.f, a1 = 0.f;
#pragma unroll
    for (int lvl = 0; lvl < 4; ++lvl) {
      const int H = Hs[lvl], W = Hs[lvl];
      const float dz = sz - glog[lvl];
      const float wl = 1.0f / (1.0f + __expf(dz * dz * 0.5f)); // sigmoid(-dz^2/TAU)
      // W*stride == 1024 for every level -> ix = sx/stride - 0.5
      const float ix = sx / strd[lvl] - 0.5f;
      const float iy = sy / strd[lvl] - 0.5f;
      const float x0f = floorf(ix), y0f = floorf(iy);
      const float wx1 = ix - x0f, wx0 = 1.f - wx1;
      const float wy1 = iy - y0f, wy0 = 1.f - wy1;
      const int x0 = (int)x0f, y0 = (int)y0f;
      const float* vt = vts[lvl];
#pragma unroll
      for (int cnr = 0; cnr < 4; ++cnr) {
        const int xs = x0 + (cnr & 1);
        const int ys = y0 + (cnr >> 1);
        const float wc = ((cnr & 1) ? wx1 : wx0) * ((cnr >> 1) ? wy1 : wy0);
        const bool in = (xs >= 0) && (xs < W) && (ys >= 0) && (ys < H);
        const float w = in ? wl * wc : 0.f;
        const int xi = min(max(xs, 0), W - 1);
        const int yi = min(max(ys, 0), H - 1);
        const v2f v = *(const v2f*)(vt + (((size_t)b * H + yi) * W + xi) * 256
                                       + g * 64 + 2 * lane);
        a0 += w * v.x;
        a1 += w * v.y;
      }
    }
    const float sav = sa[(size_t)bl * 128 + idx];
    v2f out;
    out.x = a0 * cav0 * sav;
    out.y = a1 * cav1 * sav;
    *(v2f*)(vmod + (size_t)bl * 8192 + idx * 64 + 2 * lane) = out;
  }
}

// ---------------------------------------------------------------------------
extern "C" void kernel_launch(void* const* d_in, const int* in_sizes, int n_in,
                              void* d_out, int out_size, void* d_ws, size_t ws_size,
                              hipStream_t stream) {
  (void)in_sizes; (void)n_in; (void)out_size; (void)ws_size;
  const float* v0   = (const float*)d_in[0];
  const float* v1   = (const float*)d_in[1];
  const float* v2   = (const float*)d_in[2];
  const float* v3   = (const float*)d_in[3];
  const float* qc   = (const float*)d_in[4];   // (2,1000,256)
  const float* qx   = (const float*)d_in[5];   // (2,1000,5)
  // d_in[6] = strides (known constants {4,8,16,32}, baked in)
  const float* Woff = (const float*)d_in[7];
  const float* boff = (const float*)d_in[8];
  const float* Wca1 = (const float*)d_in[9];
  const float* Wca2 = (const float*)d_in[10];
  const float* Wsa1 = (const float*)d_in[11];
  const float* Wsa2 = (const float*)d_in[12];
  const float* Wout = (const float*)d_in[13];
  const float* bout = (const float*)d_in[14];

  // workspace carve-up (floats); total ~= 251 MB
  float* ws = (float*)d_ws;
  size_t o = 0;
  float* vt0  = ws + o; o += (size_t)2 * 256 * 256 * 256;
  float* vt1  = ws + o; o += (size_t)2 * 128 * 128 * 256;
  float* vt2  = ws + o; o += (size_t)2 *  64 *  64 * 256;
  float* vt3  = ws + o; o += (size_t)2 *  32 *  32 * 256;
  float* offs = ws + o; o += (size_t)NBL * 384;
  float* hca  = ws + o; o += (size_t)NBL * 64;
  float* hsa  = ws + o; o += (size_t)NBL * 64;
  float* cab  = ws + o; o += (size_t)NBL * 256;
  float* sab  = ws + o; o += (size_t)NBL * 128;
  float* vmod = ws + o; o += (size_t)NBL * 8192;

  // 1) layout transform: CHW -> HWC per level (streaming, HBM-rate)
  transpose_chw_hwc<<<dim3(256 * 256, BQ), 256, 0, stream>>>(v0, vt0, 256 * 256);
  transpose_chw_hwc<<<dim3(128 * 128, BQ), 256, 0, stream>>>(v1, vt1, 128 * 128);
  transpose_chw_hwc<<<dim3( 64 *  64, BQ), 256, 0, stream>>>(v2, vt2,  64 *  64);
  transpose_chw_hwc<<<dim3( 32 *  32, BQ), 256, 0, stream>>>(v3, vt3,  32 *  32);

  // 2) projections (exact fp32 WMMA); grid.y covers N in 64-col groups
  wmma_gemm_f32<<<dim3(125, 6), 128, 0, stream>>>(qc, Woff, boff, nullptr, offs, NBL, 384, 256, 0);
  wmma_gemm_f32<<<dim3(125, 1), 128, 0, stream>>>(qc, Wca1, nullptr, nullptr, hca, NBL,  64, 256, 1);
  wmma_gemm_f32<<<dim3(125, 1), 128, 0, stream>>>(qc, Wsa1, nullptr, nullptr, hsa, NBL,  64, 256, 1);
  wmma_gemm_f32<<<dim3(125, 4), 128, 0, stream>>>(hca, Wca2, nullptr, nullptr, cab, NBL, 256,  64, 2);
  wmma_gemm_f32<<<dim3(125, 2), 128, 0, stream>>>(hsa, Wsa2, nullptr, nullptr, sab, NBL, 128,  64, 2);

  // 3) oriented bilinear sampling + ca/sa modulation (L2-resident gathers)
  sample_modulate<<<dim3(NBL), 128, 0, stream>>>(vt0, vt1, vt2, vt3, qx, offs, cab, sab, vmod);

  // 4) output projection, fused bias + residual: d_out = vmod @ W_out + b_out + qc
  wmma_gemm_f32<<<dim3(125, 4), 128, 0, stream>>>(vmod, Wout, bout, qc, (float*)d_out,
                                                  NBL, 256, 8192, 0);
}